// TorchDFA_74096775791262
// MI455X (gfx1250) — compile-verified
//
#include <hip/hip_runtime.h>
#include <stdint.h>

// Problem constants (from reference): D DFAs, S states, A symbols, B strings, L length
#define D_N 256
#define S_N 64
#define A_N 16
#define B_N 4096
#define L_N 1024

#define G_DFA 16            // DFAs per block
#define G_B   16            // strings per block
#define TT    128           // timesteps per staged symbol tile
#define NTILES (L_N / TT)
#define TBL_STRIDE 1088     // padded bytes per packed DFA table (1024 + 64) to spread LDS banks

// ---- gfx1250 async global->LDS path (ASYNCcnt), with safe fallback ----
#if defined(__has_builtin)
#if __has_builtin(__builtin_amdgcn_global_load_async_to_lds_b128)
#define HAVE_ASYNC_B128 1
#endif
#if __has_builtin(__builtin_amdgcn_s_wait_asynccnt)
#define HAVE_WAIT_ASYNC 1
#endif
#endif

typedef int v4i __attribute__((vector_size(16)));
typedef __attribute__((address_space(1))) v4i* gptr_v4i;
typedef __attribute__((address_space(3))) v4i* lptr_v4i;

__device__ __forceinline__ void async_cp16(void* lds_dst, const void* gsrc) {
#ifdef HAVE_ASYNC_B128
  __builtin_amdgcn_global_load_async_to_lds_b128(
      (gptr_v4i)gsrc, (lptr_v4i)lds_dst,
      /*imm offset*/ 0, /*cpol*/ 0);
#else
  *(int4*)lds_dst = *(const int4*)gsrc;
#endif
}

__device__ __forceinline__ void async_wait_all() {
#ifdef HAVE_ASYNC_B128
#ifdef HAVE_WAIT_ASYNC
  __builtin_amdgcn_s_wait_asynccnt(0);
#else
  asm volatile("s_wait_asynccnt 0" ::: "memory");
#endif
#endif
}

// ---- kernel 1: pack transition table int32 -> u8 (next state < 64 fits in a byte) ----
__global__ __launch_bounds__(256) void dfa_pack_tf(const int* __restrict__ tf,
                                                   uint8_t* __restrict__ out) {
  int i = (blockIdx.x * 256 + threadIdx.x) * 4;
  int4 v = *(const int4*)(tf + i);
  uchar4 p;
  p.x = (uint8_t)v.x; p.y = (uint8_t)v.y; p.z = (uint8_t)v.z; p.w = (uint8_t)v.w;
  *(uchar4*)(out + i) = p;
}

// ---- kernel 2: DFA simulation. One thread per (dfa, string) chain. ----
__global__ __launch_bounds__(256) void dfa_run(const int* __restrict__ x,
                                               const int* __restrict__ init_state,
                                               const uint8_t* __restrict__ accepting,
                                               const uint8_t* __restrict__ tfp,
                                               float* __restrict__ out) {
  __shared__ __align__(16) uint8_t s_tbl[G_DFA * TBL_STRIDE];       // ~17 KB
  __shared__ __align__(16) int     s_x[2][G_B * TT];                // 16 KB double buffer

  const int tid = threadIdx.x;
  const int dl  = tid & (G_DFA - 1);   // lanes 0..15 of each half-wave: one DFA each
  const int bl  = tid >> 4;            // string within block
  const int d0  = blockIdx.y * G_DFA;
  const int b0  = blockIdx.x * G_B;

  // Stage 16 packed tables: 16 KB = 1024 x 16B chunks, 4 per thread (async -> LDS)
#pragma unroll
  for (int c = 0; c < 4; ++c) {
    int k      = tid + c * 256;
    int tbi    = k >> 6;               // table index 0..15 (64 chunks per 1KB table)
    int within = (k & 63) << 4;
    async_cp16(&s_tbl[tbi * TBL_STRIDE + within],
               tfp + (d0 + tbi) * (S_N * A_N) + within);
  }
  // Stage symbol tile 0: 16 rows x 512B = 512 x 16B chunks, 2 per thread
#pragma unroll
  for (int c = 0; c < 2; ++c) {
    int k      = tid + c * 256;
    int row    = k >> 5;
    int within = (k & 31) << 4;
    async_cp16((uint8_t*)&s_x[0][row * TT] + within,
               (const uint8_t*)(x + (b0 + row) * L_N) + within);
  }
  async_wait_all();
  __syncthreads();

  int state = init_state[d0 + dl];
  const uint8_t* __restrict__ tb = &s_tbl[dl * TBL_STRIDE];

  for (int tile = 0; tile < NTILES; ++tile) {
    // Prefetch next tile into the other buffer while we consume this one.
    if (tile + 1 < NTILES) {
      const int t0n = (tile + 1) * TT;
#pragma unroll
      for (int c = 0; c < 2; ++c) {
        int k      = tid + c * 256;
        int row    = k >> 5;
        int within = (k & 31) << 4;
        async_cp16((uint8_t*)&s_x[(tile + 1) & 1][row * TT] + within,
                   (const uint8_t*)(x + (b0 + row) * L_N + t0n) + within);
      }
    }
    const int* __restrict__ xr = &s_x[tile & 1][bl * TT];
#pragma unroll 4
    for (int t = 0; t < TT; t += 4) {
      int4 s4 = *(const int4*)(xr + t);  // ds_load_b128, broadcast across 16 lanes
      state = tb[(state << 4) | (s4.x & 15)];
      state = tb[(state << 4) | (s4.y & 15)];
      state = tb[(state << 4) | (s4.z & 15)];
      state = tb[(state << 4) | (s4.w & 15)];
    }
    async_wait_all();   // our async writes into the other buffer have landed
    __syncthreads();    // everyone done reading old buffer / writes visible
  }

  // accepting_states is a JAX bool array: 1 byte per element
  out[(d0 + dl) * B_N + (b0 + bl)] =
      accepting[(d0 + dl) * S_N + state] ? 1.0f : 0.0f;
}

extern "C" void kernel_launch(void* const* d_in, const int* in_sizes, int n_in,
                              void* d_out, int out_size, void* d_ws, size_t ws_size,
                              hipStream_t stream) {
  const int*     x    = (const int*)d_in[0];      // (B, L) int32
  const int*     init = (const int*)d_in[1];      // (D,)   int32
  const uint8_t* acc  = (const uint8_t*)d_in[2];  // (D, S) bool (1 byte/elt)
  const int*     tf   = (const int*)d_in[3];      // (D, S, A) int32
  float*         out  = (float*)d_out;            // (D, B) as float 0/1
  uint8_t*       tfp  = (uint8_t*)d_ws;           // packed tables, 256 KB

  (void)in_sizes; (void)n_in; (void)out_size; (void)ws_size;

  // 262144 elements, 4 per thread
  dfa_pack_tf<<<(D_N * S_N * A_N) / (256 * 4), 256, 0, stream>>>(tf, tfp);

  dim3 grid(B_N / G_B, D_N / G_DFA);
  dfa_run<<<grid, 256, 0, stream>>>(x, init, acc, tfp, out);
}